// WLNCHL_40939628265745
// MI455X (gfx1250) — compile-verified
//
#include <hip/hip_runtime.h>

// Persistent single-WGP recurrent kernel for the structured-plasticity RNN.
// Dense dw matvec collapsed algebraically to rank-structured gather/scatter.
// Both gathers run on V_WMMA_F32_16X16X4_F32:
//   - layer-2 gather V = [u1;u2] @ h1            (phase D, 16 WMMA/wave)
//   - next-step layer-1 scalar gather (2x5 outer) (phase E,  8 WMMA/wave)
// 4 barriers/step; all reductions deterministic fixed-order.
// h2 window lives in LDS (per-thread ring row) to stay under the VGPR cap.

#define T_STEPS 256
#define N0 1024
#define N1 2048
#define N2 1024
#define WIN 16
#define H1S 17          // padded h1 row stride (17 coprime with 64 banks)
#define H2S 18          // h2 row stride: 8B-aligned b64 rows, conflict-free
#define NTHREADS 1024
#define NWAVES 32
#define VPS 33          // padded vpart per-wave stride
#define F1PS 13         // padded f1part per-wave stride

// LDS layout (float indices)
#define OFF_H1    0                        // [N1][H1S]          = 34816
#define OFF_H2    (OFF_H1 + N1*H1S)        // [N2][H2S]          = 18432
#define OFF_UU    (OFF_H2 + N2*H2S)        // [N1][2] u1/u2      = 4096
#define OFF_XX    (OFF_UU + 2*N1)          // [N0][2] {a, a*E0}  = 2048
#define OFF_Y4    (OFF_XX + 2*N0)          // [N0][4] y-columns  = 4096
#define OFF_ZERO  (OFF_Y4 + 4*N0)          // 8 zero floats (dead-lane reads)
#define OFF_VPART (OFF_ZERO + 8)           // [32][VPS]          = 1056
#define OFF_VFIN  (OFF_VPART + NWAVES*VPS) // [2][16]            = 32
#define OFF_F1P   (OFF_VFIN + 32)          // [32][F1PS]         = 416
#define OFF_F1    (OFF_F1P + NWAVES*F1PS)  // [10] (pad 16)
#define OFF_S2P   (OFF_F1 + 16)            // [32][6]            = 192
#define OFF_F2    (OFF_S2P + 192)          // [6]  (pad 16)
#define LDS_FLOATS (OFF_F2 + 16)           // ~255 KB of the 320 KB WGP LDS

typedef float v2f __attribute__((ext_vector_type(2)));
typedef float v4f __attribute__((ext_vector_type(4)));
typedef float v8f __attribute__((ext_vector_type(8)));

#if __has_builtin(__builtin_amdgcn_tanhf)
#define TANHF(x) __builtin_amdgcn_tanhf(x)   // v_tanh_f32, 1 TRANS op
#else
#define TANHF(x) tanhf(x)
#endif

__device__ __forceinline__ float wave_sum32(float v) {
#pragma unroll
  for (int m = 16; m >= 1; m >>= 1) v += __shfl_xor(v, m, 32);
  return v;
}

// 8-WMMA gather of next step's 10 layer-1 scalars:
//   rows {a, a*E0} (xx), cols {C*ap, C*a, 1, A*sin, D} (y4 + const-1 column)
__device__ __forceinline__ void f1_gather(const float* xx, const float* y4,
                                          float* f1p, int wave, int lane,
                                          int ax_base, int ax_step,
                                          int yb_base, int yb_step, bool isOne) {
  v8f accA = {0.f,0.f,0.f,0.f,0.f,0.f,0.f,0.f};
#pragma unroll
  for (int c = 0; c < 8; ++c) {
    const int ai = ax_base + c*ax_step;
    const int yi = yb_base + c*yb_step;
    v2f av, bv;
    av[0] = xx[ai];     av[1] = xx[ai + 2];
    const float b0 = y4[yi], b1 = y4[yi + 4];
    bv[0] = isOne ? 1.f : b0;
    bv[1] = isOne ? 1.f : b1;
    accA = __builtin_amdgcn_wmma_f32_16x16x4_f32(
               false, av, false, bv, (short)0, accA, false, false);
  }
  if (lane < 5) {                  // C rows M=0 (a-weighted), M=1 (ae-weighted)
    f1p[wave*F1PS +     lane] = accA[0];
    f1p[wave*F1PS + 6 + lane] = accA[1];
  }
}

extern "C" __global__ void __launch_bounds__(NTHREADS, 1)
hebb_scan_kernel(const float* __restrict__ x,
                 const float* __restrict__ p0,
                 const float* __restrict__ p1,
                 const float* __restrict__ p2,
                 const float* __restrict__ h0_in,
                 const float* __restrict__ h1_in,
                 const float* __restrict__ h2_in,
                 float* __restrict__ out) {
  extern __shared__ float lds[];
  float* h1    = lds + OFF_H1;
  float* h2    = lds + OFF_H2;
  float* uu    = lds + OFF_UU;
  float* xx    = lds + OFF_XX;
  float* y4    = lds + OFF_Y4;
  float* vpart = lds + OFF_VPART;
  float* vfin  = lds + OFF_VFIN;
  float* f1p   = lds + OFF_F1P;
  float* f1    = lds + OFF_F1;
  float* s2p   = lds + OFF_S2P;
  float* f2    = lds + OFF_F2;

  const int t    = threadIdx.x;      // 0..1023
  const int lane = t & 31;
  const int wave = t >> 5;

  // ---------------- preload all parameters into registers ----------------
  const float A0 = p0[t*5+0], C0 = p0[t*5+2], D0 = p0[t*5+3], E0 = p0[t*5+4];
  float P1[2][5];
#pragma unroll
  for (int h = 0; h < 2; ++h)
#pragma unroll
    for (int c = 0; c < 5; ++c) P1[h][c] = p1[(t + h*N0)*5 + c];
  const float B2o = p2[t*5+1], C2o = p2[t*5+2], D2o = p2[t*5+3], E2o = p2[t*5+4];

  // ---------------- initial state ----------------
  float s1r[2];
#pragma unroll
  for (int h = 0; h < 2; ++h) {
    const int j = t + h*N0;
    float s = 0.f;
    for (int k = 0; k < WIN; ++k) {
      float v = h1_in[j*WIN + k];
      h1[j*H1S + k] = v;              // ring start = 0: logical == physical
      s += v;
    }
    s1r[h] = s;
  }
#pragma unroll
  for (int k = 0; k < WIN; ++k) h2[t*H2S + k] = h2_in[t*WIN + k];
  const float ap0 = h0_in[t*WIN + 1];
  float hcon0 = 0.f;                  // constant tail of h0 row sum (cols 2..15)
  for (int k = 2; k < WIN; ++k) hcon0 += h0_in[t*WIN + k];
  if (t < 8) lds[OFF_ZERO + t] = 0.f; // persistent zero slot (dead-lane reads)

  // a0 pipeline (depends only on x -> computable ahead of the recurrence)
  float aC = TANHF(x[t]);             // a(0)
  float aN = TANHF(x[N0 + t]);        // a(1)
  float xv = x[2*N0 + t];             // x for a(2)

  // prime gather operands for step 0 (a = aC, ap = h0 init)
  { v2f xv2; xv2[0] = aC; xv2[1] = aC*E0; *(v2f*)(xx + 2*t) = xv2;
    v4f yv;  yv[0] = C0*ap0; yv[1] = C0*aC;
             yv[2] = A0*(ap0 + aC + hcon0); yv[3] = D0;
    *(v4f*)(y4 + 4*t) = yv; }
  __syncthreads();

  // ---------- per-lane WMMA operand addressing (branchless builds) ----------
  const int  r   = lane & 15;         // A-matrix row / B-matrix column
  const int  hi  = lane >> 4;         // K-pair select {0,1} vs {2,3}
  const bool ract = (r < 2);
  // phase-D (layer-2 gather): A from uu, B from h1
  const int a_base = ract ? (wave*128 + hi*4 + r) : (OFF_ZERO - OFF_UU);
  const int a_step = ract ? 8 : 0;
  const int b_base = wave*64*H1S + hi*2*H1S + r;
  // phase-E (f1 gather): A from xx, B from y4 (+ constant-1 column at n=2)
  const int ax_base = ract ? (wave*64 + hi*4 + r) : (OFF_ZERO - OFF_XX);
  const int ax_step = ract ? 8 : 0;
  const bool yact   = ract || r == 3 || r == 4;
  const int  ysel   = ract ? r : (r == 3 ? 2 : 3);
  const int  yb_base = yact ? (wave*128 + hi*8 + ysel) : (OFF_ZERO - OFF_Y4);
  const int  yb_step = yact ? 16 : 0;
  const bool isOne   = (r == 2);

  // priming pass: f1part for step 0
  f1_gather(xx, y4, f1p, wave, lane, ax_base, ax_step, yb_base, yb_step, isOne);
  __syncthreads();

  // ================= time loop (sequential recurrence) =================
  for (int ts = 0; ts < T_STEPS; ++ts) {
    const float xnext = (ts + 3 < T_STEPS) ? x[(ts + 3)*N0 + t] : 0.f; // prefetch

    // ---- Phase A: f1 finalize (10 waves) + look-ahead a0 + gather operand store
    if (t < 10*32) {
      const int q = wave;                      // f1[q] = row(q>=5) x col(q%5)
      const int rr = (q >= 5) ? 1 : 0, cc = q - 5*rr;
      float s = wave_sum32(f1p[lane*F1PS + rr*6 + cc]);
      if (lane == 0) f1[q] = s;
    }
    const float a_new = TANHF(xv);             // a(ts+2)
    { // operands for step ts+1's f1 gather: a = aN, ap = aC
      v2f xv2; xv2[0] = aN; xv2[1] = aN*E0; *(v2f*)(xx + 2*t) = xv2;
      v4f yv;  yv[0] = C0*aC; yv[1] = C0*aN;
               yv[2] = A0*(aC + aN + hcon0); yv[3] = D0;
      *(v4f*)(y4 + 4*t) = yv; }
    __syncthreads();                                  // B1

    // ---- Phase C: layer-1 scatter (a1, h1 append) + layer-2 gather prep ----
    {
      const int pAa = ts & 15, pBb = (ts + 1) & 15;   // logical 0,1 of h1 ring
      const float v20=f1[0], v21=f1[1], sA =f1[2], c3b=f1[3], c4b=f1[4];
      const float v10=f1[5], v11=f1[6], sAe=f1[7], c3a=f1[8], c4a=f1[9];
      float q2[6] = {0,0,0,0,0,0};
#pragma unroll
      for (int h = 0; h < 2; ++h) {
        const int j = t + h*N0;
        const float Ai=P1[h][0], Bo=P1[h][1], Co=P1[h][2], Do=P1[h][3], Eo=P1[h][4];
        float s1 = s1r[h];
        const float hl0 = h1[j*H1S + pAa];
        const float hl1 = h1[j*H1S + pBb];
        const float dot = (v10 + Eo*v20)*hl0 + (v11 + Eo*v21)*hl1;
        const float o = 0.5f*( Co*dot + Bo*s1*(sAe + Eo*sA)
                             + (c3a + Eo*c3b) + (float)WIN*Do*(c4a + Eo*c4b) );
        const float a1 = TANHF(o);
        h1[j*H1S + pAa] = a1;                         // append into ring slot
        s1 += a1 - hl0;
        s1r[h] = s1;
        const float ae1 = a1 * Eo;
        v2f uv; uv[0] = ae1 * Co; uv[1] = a1 * Co;    // u1,u2 interleaved
        *(v2f*)(uu + 2*j) = uv;
        q2[0] += ae1;          q2[1] += a1;
        q2[2] += ae1*Ai*s1;    q2[3] += a1*Ai*s1;     // s_in = s1 post-append
        q2[4] += ae1*Do;       q2[5] += a1*Do;
      }
#pragma unroll
      for (int q = 0; q < 6; ++q) {
        float s = wave_sum32(q2[q]);
        if (lane == 0) s2p[wave*6 + q] = s;
      }
    }
    __syncthreads();                                  // B3

    // ---- Phase D: WMMA gather V = [u1;u2] @ h1 (K=2048, 16 WMMA/wave) ----
    {
      v8f acc0 = {0.f,0.f,0.f,0.f,0.f,0.f,0.f,0.f};
      v8f acc1 = acc0;
#pragma unroll
      for (int c = 0; c < 16; ++c) {
        const int aidx = a_base + c*a_step;
        const int bidx = b_base + c*4*H1S;
        v2f av, bv;
        av[0] = uu[aidx];  av[1] = uu[aidx + 2];
        bv[0] = h1[bidx];  bv[1] = h1[bidx + H1S];
        if (c & 1)
          acc1 = __builtin_amdgcn_wmma_f32_16x16x4_f32(
                     false, av, false, bv, (short)0, acc1, false, false);
        else
          acc0 = __builtin_amdgcn_wmma_f32_16x16x4_f32(
                     false, av, false, bv, (short)0, acc0, false, false);
      }
      if (lane < 16) {
        vpart[wave*VPS +      lane] = acc0[0] + acc1[0];   // v1 partials
        vpart[wave*VPS + 16 + lane] = acc0[1] + acc1[1];   // v2 partials
      }
      if (t < 6*32) {                  // finalize layer-2 scalars (6 waves)
        const int q = wave;
        float s = wave_sum32(s2p[lane*6 + q]);
        if (lane == 0) f2[q] = s;
      }
    }
    __syncthreads();                                  // B4

    // ---- Phase E: next-step f1 gather (8 WMMA/wave) + vpart reduce ----
    f1_gather(xx, y4, f1p, wave, lane, ax_base, ax_step, yb_base, yb_step, isOne);
    {
      const int rr = wave >> 4, pc = wave & 15;       // 32 waves = 32 outputs
      float s = wave_sum32(vpart[lane*VPS + rr*16 + pc]);
      if (lane == 0) {
        // store rotated into h2-PHYSICAL index: q = (ts + lc) & 15 = (pc-1) & 15
        const int q = (pc - 1) & 15;
        vfin[rr*16 + q] = s;
      }
    }
    __syncthreads();                                  // B5

    // ---- Phase F: layer-2 scatter (h2 ring row in LDS, physically indexed) ----
    {
      float hh[WIN];
#pragma unroll
      for (int c2 = 0; c2 < WIN/2; ++c2) {
        v2f hv = *(const v2f*)(h2 + t*H2S + 2*c2);    // conflict-free b64 loads
        hh[2*c2] = hv[0]; hh[2*c2+1] = hv[1];
      }
      float dot = 0.f, s2 = 0.f;
#pragma unroll
      for (int k = 0; k < WIN; ++k) {
        dot += (vfin[k] + E2o*vfin[16 + k]) * hh[k];  // both h2-physical order
        s2  += hh[k];                                 // window sum, pre-append
      }
      const float o = 0.5f*( C2o*dot + B2o*s2*(f2[0] + E2o*f2[1])
                           + (f2[2] + E2o*f2[3])
                           + (float)WIN*D2o*(f2[4] + E2o*f2[5]) );
      const float a2 = TANHF(o);
      out[ts*N2 + t] = a2;
      h2[t*H2S + (ts & 15)] = a2;                     // append into ring slot
    }
    aC = aN; aN = a_new; xv = xnext;   // rotate look-ahead pipeline
  }
}

extern "C" void kernel_launch(void* const* d_in, const int* in_sizes, int n_in,
                              void* d_out, int out_size, void* d_ws, size_t ws_size,
                              hipStream_t stream) {
  (void)in_sizes; (void)n_in; (void)d_ws; (void)ws_size; (void)out_size;
  const float* x  = (const float*)d_in[0];
  const float* p0 = (const float*)d_in[1];
  const float* p1 = (const float*)d_in[2];
  const float* p2 = (const float*)d_in[3];
  const float* h0 = (const float*)d_in[4];
  const float* h1 = (const float*)d_in[5];
  const float* h2 = (const float*)d_in[6];
  float* out = (float*)d_out;

  const size_t smem = (size_t)LDS_FLOATS * sizeof(float);   // ~255 KB
  (void)hipFuncSetAttribute((const void*)hebb_scan_kernel,
                            hipFuncAttributeMaxDynamicSharedMemorySize, (int)smem);
  hebb_scan_kernel<<<1, NTHREADS, smem, stream>>>(x, p0, p1, p2, h0, h1, h2, out);
}